// DGLAttentionModule_46566035423801
// MI455X (gfx1250) — compile-verified
//
#include <hip/hip_runtime.h>
#include <hip/hip_bf16.h>

// ---------------------------------------------------------------------------
// Problem constants (match reference)
// ---------------------------------------------------------------------------
#define N_NODES   10000
#define N_EDGES   320000
#define DIM       256
#define NUM_HEADS 8
#define HEAD_DIM  32

typedef __attribute__((ext_vector_type(2))) float v2f;
typedef __attribute__((ext_vector_type(8))) float v8f;

// ---------------------------------------------------------------------------
// Y = X @ W^T + b   (X: [M,256] row-major, W: [256,256] row-major, Y: [M,256])
// One 16x16 output tile per wave32 via V_WMMA_F32_16X16X4_F32.
//   A frag (16x4 f32): lane L (0..15) = row M=L, VGPR g holds K = g + (L>>4)*2
//   B frag (4x16 f32): lane L, VGPR g holds K = g + (L>>4)*2, N = L&15
//   C/D  (16x16 f32):  lane L, VGPR g holds M = g + (L>>4)*8, N = L&15
// Both A and B fragments are contiguous float2 loads under this layout.
// ---------------------------------------------------------------------------
#define XS_STRIDE 264  // 256 + 8 pad, keeps rows 16B-aligned, spreads LDS banks

__global__ __launch_bounds__(256) void gemm_xwT(const float* __restrict__ X,
                                                const float* __restrict__ W,
                                                const float* __restrict__ bias,
                                                float* __restrict__ Y) {
    __shared__ float xs[16 * XS_STRIDE];

    const int tid = threadIdx.x;
    const int m0  = blockIdx.x * 16;

    // Cooperatively stage the 16x256 A-strip into LDS (float4 per thread x4).
#pragma unroll
    for (int i = 0; i < 4; ++i) {
        const int f4  = tid + i * 256;     // float4 index within strip
        const int row = f4 >> 6;           // 64 float4 per row
        const int c4  = (f4 & 63) << 2;    // float column
        const float4 val = *(const float4*)(X + (m0 + row) * DIM + c4);
        *(float4*)(&xs[row * XS_STRIDE + c4]) = val;
    }
    __syncthreads();

    const int wave = tid >> 5;
    const int lane = tid & 31;
    const int n0   = (blockIdx.y * 8 + wave) * 16;
    const int nn   = lane & 15;            // N within tile / M row for A frag
    const int koff = (lane >> 4) << 1;     // 0 or 2: K sub-offset per half-wave

    // Bias depends only on N -> identical across all 8 C registers.
    const float bval = bias[n0 + nn];
    v8f c;
#pragma unroll
    for (int i = 0; i < 8; ++i) c[i] = bval;

    const float* wrow = W + (n0 + nn) * DIM + koff;   // B frag source
    const float* arow = &xs[nn * XS_STRIDE + koff];   // A frag source (LDS)

#pragma unroll 8
    for (int k0 = 0; k0 < DIM; k0 += 4) {
        const v2f a  = *(const v2f*)(arow + k0);
        const v2f bf = *(const v2f*)(wrow + k0);
        // 8 args: (neg_a, A, neg_b, B, c_mod, C, reuse_a, reuse_b)
        c = __builtin_amdgcn_wmma_f32_16x16x4_f32(false, a, false, bf,
                                                  (short)0, c, false, false);
    }

    // Store D: lane L, VGPR g -> row m0 + g + (L>>4)*8, col n0 + (L&15)
    const int mrow = m0 + ((lane >> 4) << 3);
#pragma unroll
    for (int g = 0; g < 8; ++g) {
        Y[(mrow + g) * DIM + n0 + nn] = c[g];
    }
}

// ---------------------------------------------------------------------------
// One wave32 per edge: 8 head dot-products (lane = head-dim element),
// softmax over the HEAD axis, weighted v[src] scatter-added into h[dst].
// q/k/v/h tables are L2-resident (10.2 MB each << 192 MB L2).
// ---------------------------------------------------------------------------
__global__ __launch_bounds__(256) void edge_attn(const float* __restrict__ q,
                                                 const float* __restrict__ k,
                                                 const float* __restrict__ v,
                                                 const int* __restrict__ src,
                                                 const int* __restrict__ dst,
                                                 float* __restrict__ h,
                                                 int E) {
    const int e = (blockIdx.x * blockDim.x + threadIdx.x) >> 5;
    if (e >= E) return;
    const int lane = threadIdx.x & 31;

    const int s = src[e];
    const int d = dst[e];
    const float* qr = q + s * DIM;
    const float* kr = k + d * DIM;
    const float* vr = v + s * DIM;

    // Warm the v row while we compute scores (global_prefetch_b8 on gfx1250).
    __builtin_prefetch(vr + lane * 8, 0, 1);

    float score[NUM_HEADS];
#pragma unroll
    for (int hh = 0; hh < NUM_HEADS; ++hh) {
        float p = qr[hh * HEAD_DIM + lane] * kr[hh * HEAD_DIM + lane];
        // wave32 butterfly reduction -> every lane holds the head dot product
#pragma unroll
        for (int m = 16; m >= 1; m >>= 1) p += __shfl_xor(p, m, 32);
        score[hh] = p * 0.17677669529663687f;  // 1/sqrt(HEAD_DIM)
    }

    // softmax over the heads axis (reference: softmax(dim=1) on [E,H])
    float mx = score[0];
#pragma unroll
    for (int hh = 1; hh < NUM_HEADS; ++hh) mx = fmaxf(mx, score[hh]);
    float sum = 0.0f;
#pragma unroll
    for (int hh = 0; hh < NUM_HEADS; ++hh) {
        score[hh] = __expf(score[hh] - mx);
        sum += score[hh];
    }
    const float inv = __frcp_rn(sum);

    float* hd = h + d * DIM;
#pragma unroll
    for (int hh = 0; hh < NUM_HEADS; ++hh) {
        const float val = vr[hh * HEAD_DIM + lane] * (score[hh] * inv);
        atomicAdd(&hd[hh * HEAD_DIM + lane], val);  // global_atomic_add_f32
    }
}

// ---------------------------------------------------------------------------
__global__ __launch_bounds__(256) void zero_f32(float* __restrict__ p, int n) {
    const int i = blockIdx.x * blockDim.x + threadIdx.x;
    if (i < n) p[i] = 0.0f;
}

// ---------------------------------------------------------------------------
extern "C" void kernel_launch(void* const* d_in, const int* in_sizes, int n_in,
                              void* d_out, int out_size, void* d_ws, size_t ws_size,
                              hipStream_t stream) {
    const float* x   = (const float*)d_in[0];
    const int*   src = (const int*)d_in[1];
    const int*   dst = (const int*)d_in[2];
    const float* Wq  = (const float*)d_in[3];
    const float* bq  = (const float*)d_in[4];
    const float* Wk  = (const float*)d_in[5];
    const float* bk  = (const float*)d_in[6];
    const float* Wv  = (const float*)d_in[7];
    const float* bv  = (const float*)d_in[8];
    const float* Wo  = (const float*)d_in[9];
    const float* bo  = (const float*)d_in[10];
    float* out = (float*)d_out;

    const int NF = N_NODES * DIM;  // 2,560,000 floats per table
    float* q  = (float*)d_ws;
    float* kk = q + NF;
    float* vv = kk + NF;
    float* hh = vv + NF;

    // 625 M-tiles x (2 x 8 waves) N-tiles covers [10000, 256] exactly.
    dim3 gg(N_NODES / 16, 2, 1);
    gemm_xwT<<<gg, 256, 0, stream>>>(x, Wq, bq, q);
    gemm_xwT<<<gg, 256, 0, stream>>>(x, Wk, bk, kk);
    gemm_xwT<<<gg, 256, 0, stream>>>(x, Wv, bv, vv);

    zero_f32<<<(NF + 255) / 256, 256, 0, stream>>>(hh, NF);

    // 320000 edges = 40000 blocks x 8 waves, exact.
    edge_attn<<<(N_EDGES * 32) / 256, 256, 0, stream>>>(q, kk, vv, src, dst, hh, N_EDGES);

    gemm_xwT<<<gg, 256, 0, stream>>>(hh, Wo, bo, out);

    (void)in_sizes; (void)n_in; (void)out_size; (void)ws_size;
}